// InitialVQVAE_29600914604836
// MI455X (gfx1250) — compile-verified
//
#include <hip/hip_runtime.h>
#include <hip/hip_bf16.h>

typedef __attribute__((ext_vector_type(2))) float v2f;
typedef __attribute__((ext_vector_type(8))) float v8f;

#define BATCH   8
#define OUTN    6291456      // 8*3*16*128*128
#define NPOS    262144       // 8*8*64*64  (VQ positions)
#define ZC      8
#define MCODES  32
#define ZTOT    2097152      // 8*8*8*64*64

#if defined(__has_builtin)
# if __has_builtin(__builtin_amdgcn_global_load_async_to_lds_b32)
#  define HAVE_ASYNC_LDS 1
# endif
#endif

typedef __attribute__((address_space(1))) int g_i32;
typedef __attribute__((address_space(3))) int l_i32;

static __device__ __forceinline__ void wait_async_lds() {
#if defined(HAVE_ASYNC_LDS)
# if __has_builtin(__builtin_amdgcn_s_wait_asynccnt)
    __builtin_amdgcn_s_wait_asynccnt(0);
# else
    asm volatile("s_wait_asynccnt 0x0" ::: "memory");
# endif
#endif
}

// D(16x16) = A(16x4) * B(4x16) + C, fp32, wave32.
static __device__ __forceinline__ v8f wmma4(v2f a, v2f b, v8f c) {
    return __builtin_amdgcn_wmma_f32_16x16x4_f32(
        false, a, false, b, (short)0, c, false, false);
}

// ---------------------------------------------------------------------------
// Implicit-GEMM 3D conv / transposed-conv via WMMA f32 16x16x4.
// One launch covers ONE batch element (src/dst pre-offset on host).
// The block stages ONE shared input tile covering WAVES output rows; wave i
// computes output row oh0+i (16 consecutive w outputs x NT*16 channels).
//   CIP : input channels padded to K-group size (4/8/32, power of 2)
//   CIN : real input channels ; COUTR : real output channels
//   NT  : 16-wide N tiles (1 or 2) ; SR : input stride (1 or 2)
//   UPS : stride-2 transposed conv in gather form (zero-stuffed input grid)
//   ACT : 0 = ReLU, 1 = tanh ; TRW : ConvT weights (in,out,k^3), flipped
// ---------------------------------------------------------------------------
template<int CIP, int CIN, int COUTR, int NT, int SR, bool UPS, int ACT,
         bool TRW, int DI, int HI, int WI, int DO, int HO, int WO,
         int WAVES, int TPW>
__global__ __launch_bounds__(WAVES * 32)
void conv_wmma_kernel(const float* __restrict__ src,
                      const float* __restrict__ wsrc,
                      const float* __restrict__ bias,
                      float* __restrict__ dst)
{
    constexpr int WP    = (SR == 2) ? 36 : 20;       // padded LDS tile width
    constexpr int WREAL = (SR == 2) ? 33 : 18;       // real cols needed
    constexpr int ROWS  = (WAVES - 1) * SR + 3;      // shared input h-rows
    constexpr int NCOL  = 16 * NT;
    constexpr int KTOT  = 27 * CIP;
    constexpr int STEPS = KTOT / 4;
    constexpr int WT    = WO / 16;
    constexpr int NTH   = HO / WAVES;                // h-blocks
    constexpr int NTILES = DO * NTH * WT;            // per-batch block tiles
    constexpr int TSLOTS = CIP * 3 * ROWS * WP;

    __shared__ float w_lds[KTOT * NCOL];
    __shared__ float in_lds[TSLOTS];

    const int tid  = threadIdx.x;
    const int lane = tid & 31;
    const int wave = tid >> 5;
    const int m    = lane & 15;       // M row (output w within tile)
    const int n    = lane & 15;       // N col (output channel)
    const int half = lane >> 4;
    const int koff = half * 2;        // K sub-offset for A/B fragments

    // ---- stage weights into LDS (block-cooperative, once) -----------------
    for (int e = tid; e < KTOT * NCOL; e += WAVES * 32) {
        int co  = e % NCOL;            // power-of-2 modulo
        int K   = e / NCOL;
        int ci  = K % CIP;
        int tap = K / CIP;
        float w = 0.f;
        if (co < COUTR && ci < CIN) {
            int widx = TRW ? ((ci * COUTR + co) * 27 + (26 - tap))  // flip
                           : ((co * CIN  + ci) * 27 + tap);
            w = wsrc[widx];
        }
        w_lds[e] = w;
    }

    for (int t = 0; t < TPW; ++t) {
        int T = blockIdx.x * TPW + t;          // uniform across block
        if (T >= NTILES) break;
        int wt  = T % WT;
        int tmp = T / WT;
        int hb  = tmp % NTH;
        int od  = tmp / NTH;
        int oh0 = hb * WAVES;
        int w0  = wt * 16;
        int ih0 = oh0 * SR - 1;

        __syncthreads();   // tile buffer free (also covers weight staging)

        // ---- stage shared input tile: [ci][kd(3)][rr(ROWS)][WP] -----------
        for (int e = tid; e < TSLOTS; e += WAVES * 32) {
            int c   = e % WP;
            int r   = e / WP;
            int rr  = r % ROWS;
            int r2  = r / ROWS;
            int kd  = r2 % 3;
            int ci  = r2 / 3;
            int id  = od * SR - 1 + kd;
            int ih  = ih0 + rr;
            int iw  = w0 * SR - 1 + c;
            bool ok = (c < WREAL) && (ci < CIN);
            int sd = id, sh = ih, sw = iw;
            if (UPS) {
                ok = ok && id >= 0 && ih >= 0 && iw >= 0 &&
                     !(id & 1) && !(ih & 1) && !(iw & 1);
                sd = id >> 1; sh = ih >> 1; sw = iw >> 1;
                ok = ok && sd < DI && sh < HI && sw < WI;
            } else {
                ok = ok && id >= 0 && id < DI && ih >= 0 && ih < HI &&
                     iw >= 0 && iw < WI;
            }
            if (ok) {
                const float* gp = src + (((size_t)ci * DI + sd) * HI + sh) * WI + sw;
#if defined(HAVE_ASYNC_LDS)
                __builtin_amdgcn_global_load_async_to_lds_b32(
                    (g_i32*)(size_t)(const void*)gp,
                    (l_i32*)&in_lds[e], 0, 0);
#else
                in_lds[e] = *gp;
#endif
            } else {
                in_lds[e] = 0.f;
            }
        }
        wait_async_lds();
        __syncthreads();

        // ---- accumulators (C initialized with bias broadcast over M) ------
        v8f acc0, acc1;
        {
            float b0 = (n < COUTR) ? bias[n] : 0.f;
            for (int r = 0; r < 8; ++r) acc0[r] = b0;
        }
        if constexpr (NT == 2) {
            float b1 = (16 + n < COUTR) ? bias[16 + n] : 0.f;
            for (int r = 0; r < 8; ++r) acc1[r] = b1;
        } else {
            for (int r = 0; r < 8; ++r) acc1[r] = 0.f;
        }

        // ---- K loop: K = tap*CIP + ci; each WMMA quad stays inside one tap
        #pragma unroll 4
        for (int s = 0; s < STEPS; ++s) {
            int ci0 = (4 * s) % CIP;       // wave-uniform
            int tap = (4 * s) / CIP;
            int kd  = tap / 9;
            int kh  = (tap % 9) / 3;
            int kw  = tap % 3;
            int ciA = ci0 + koff;
            const float* ap =
                &in_lds[((ciA * 3 + kd) * ROWS + wave * SR + kh) * WP + m * SR + kw];
            v2f a; a[0] = ap[0]; a[1] = ap[3 * ROWS * WP];   // ci -> ci+1
            const float* bp = &w_lds[(4 * s + koff) * NCOL + n];
            v2f b; b[0] = bp[0]; b[1] = bp[NCOL];            // K -> K+1
            acc0 = wmma4(a, b, acc0);
            if constexpr (NT == 2) {
                v2f b2; b2[0] = bp[16]; b2[1] = bp[NCOL + 16];
                acc1 = wmma4(a, b2, acc1);
            }
        }

        // ---- epilogue: lane holds co=n, w = w0 + half*8 + r (contiguous) --
        int oh = oh0 + wave;
        auto actf = [](float v) {
            return (ACT == 1) ? tanhf(v) : fmaxf(v, 0.f);
        };
        if (n < COUTR) {
            size_t ob = (((size_t)n * DO + od) * HO + oh) * WO + w0 + half * 8;
            float4 lo = { actf(acc0[0]), actf(acc0[1]), actf(acc0[2]), actf(acc0[3]) };
            float4 hi = { actf(acc0[4]), actf(acc0[5]), actf(acc0[6]), actf(acc0[7]) };
            *(float4*)(dst + ob)     = lo;
            *(float4*)(dst + ob + 4) = hi;
        }
        if constexpr (NT == 2) {
            if (16 + n < COUTR) {
                size_t ob = (((size_t)(16 + n) * DO + od) * HO + oh) * WO
                            + w0 + half * 8;
                float4 lo = { actf(acc1[0]), actf(acc1[1]), actf(acc1[2]), actf(acc1[3]) };
                float4 hi = { actf(acc1[4]), actf(acc1[5]), actf(acc1[6]), actf(acc1[7]) };
                *(float4*)(dst + ob)     = lo;
                *(float4*)(dst + ob + 4) = hi;
            }
        }
    }
}

// ---------------------------------------------------------------------------
// VQ kernels
// ---------------------------------------------------------------------------
__global__ void vq_init_kernel(float* stats) {
    if (threadIdx.x < 64) stats[threadIdx.x] = 0.f;   // counts[32] + sse + pad
}

__global__ __launch_bounds__(256)
void vq_kernel(const float* __restrict__ z, const float* __restrict__ emb,
               const int* __restrict__ epoch_p, float* __restrict__ zmix,
               float* __restrict__ codes_out, float* __restrict__ stats)
{
    __shared__ float e_lds[MCODES * ZC];
    __shared__ float red[256];
    __shared__ unsigned hist[MCODES];
    int tid = threadIdx.x;
    for (int i = tid; i < MCODES * ZC; i += 256) e_lds[i] = emb[i];
    if (tid < MCODES) hist[tid] = 0u;
    __syncthreads();

    int epoch  = epoch_p[0];
    bool full  = epoch >= 40;
    float alpha = (epoch <= 20) ? 0.f : (full ? 1.f : (epoch - 20) * (1.f / 20.f));

    int p = blockIdx.x * 256 + tid;
    float sse = 0.f;
    if (p < NPOS) {
        int wz = p & 63, hz = (p >> 6) & 63, dz = (p >> 12) & 7, nb = p >> 15;
        int base = nb * 262144 + dz * 4096 + hz * 64 + wz;  // channel stride 32768
        float zv[ZC];
        #pragma unroll
        for (int c = 0; c < ZC; ++c) zv[c] = z[base + c * 32768];
        int best = 0; float bestd = 3.4e38f;
        for (int j = 0; j < MCODES; ++j) {
            float d = 0.f;
            #pragma unroll
            for (int c = 0; c < ZC; ++c) {
                float t = zv[c] - e_lds[j * ZC + c];
                d += t * t;
            }
            if (d < bestd) { bestd = d; best = j; }
        }
        #pragma unroll
        for (int c = 0; c < ZC; ++c) {
            float diff = e_lds[best * ZC + c] - zv[c];
            sse += diff * diff;
            zmix[base + c * 32768] = zv[c] + alpha * diff;  // alpha=1 -> quant
        }
        codes_out[p] = full ? (float)best : 0.f;
        if (full) atomicAdd(&hist[best], 1u);
    }
    if (!full) sse = 0.f;
    red[tid] = sse;
    __syncthreads();
    for (int off = 128; off > 0; off >>= 1) {
        if (tid < off) red[tid] += red[tid + off];
        __syncthreads();
    }
    if (tid == 0) atomicAdd(&stats[32], red[0]);
    if (tid < MCODES) {
        unsigned h = hist[tid];
        if (h) atomicAdd(((unsigned*)stats) + tid, h);
    }
}

__global__ void vq_finalize_kernel(const float* __restrict__ stats,
                                   const float* __restrict__ cluster_size,
                                   const int* __restrict__ epoch_p,
                                   float* __restrict__ out)
{
    int lane = threadIdx.x;
    bool full = epoch_p[0] >= 40;
    float ent = 0.f, usedc = 0.f;
    if (lane < MCODES) {
        unsigned cnt = ((const unsigned*)stats)[lane];
        float avg = (float)cnt / (float)NPOS;
        ent = avg * logf(avg + 1e-10f);
        usedc = (cluster_size[lane] > 1e-5f) ? 1.f : 0.f;
    }
    for (int off = 16; off > 0; off >>= 1) {
        ent   += __shfl_down(ent, off, 32);
        usedc += __shfl_down(usedc, off, 32);
    }
    if (lane == 0) {
        out[OUTN]                 = full ? stats[32] / (float)ZTOT : 0.f; // vq_loss
        out[OUTN + 1 + NPOS]      = full ? expf(-ent) : 0.f;             // perplexity
        out[OUTN + 1 + NPOS + 1]  = full ? usedc / (float)MCODES : 0.f;  // used
    }
}

// ---------------------------------------------------------------------------
extern "C" void kernel_launch(void* const* d_in, const int* in_sizes, int n_in,
                              void* d_out, int out_size, void* d_ws, size_t ws_size,
                              hipStream_t stream)
{
    const float* x    = (const float*)d_in[0];
    const float* ew1  = (const float*)d_in[1];
    const float* eb1  = (const float*)d_in[2];
    const float* ew2  = (const float*)d_in[3];
    const float* eb2  = (const float*)d_in[4];
    const float* dw1  = (const float*)d_in[5];
    const float* db1  = (const float*)d_in[6];
    const float* dw2  = (const float*)d_in[7];
    const float* db2  = (const float*)d_in[8];
    const float* emb  = (const float*)d_in[9];
    const float* csz  = (const float*)d_in[10];
    const int*   epo  = (const int*)d_in[11];
    float* out = (float*)d_out;
    float* ws  = (float*)d_ws;

    // workspace: per-batch scratch for y1/y3 keeps both L2-resident between
    // producer/consumer launches. Total ~59 MB.
    float* y1b   = ws + 0;          // 2,097,152  (8*16*128*128, one batch)
    float* y3b   = ws + 2097152;    // 8,388,608  (32*16*128*128, one batch)
    float* z     = ws + 10485760;   // 2,097,152  (full batch)
    float* zmix  = ws + 12582912;   // 2,097,152  (full batch)
    float* stats = ws + 14680064;   // 64

    const size_t XSTR = 3u * 16 * 128 * 128;   // 786432
    const size_t ZSTR = 8u * 8 * 64 * 64;      // 262144

    // Encoder, batch-chunked so conv2 reads y1 out of L2 (8.4 MB chunk).
    for (int nb = 0; nb < BATCH; ++nb) {
        // conv1: 3->8, k3 s1 p1, ReLU.  tiles = 16*16*8 = 2048 -> 512 blocks
        conv_wmma_kernel<4, 3, 8, 1, 1, false, 0, false,
                         16, 128, 128, 16, 128, 128, 8, 4>
            <<<512, 256, 0, stream>>>(x + nb * XSTR, ew1, eb1, y1b);
        // conv2: 8->8, k3 s2 p1, tanh.  tiles = 8*8*4 = 256 -> 64 blocks
        conv_wmma_kernel<8, 8, 8, 1, 2, false, 1, false,
                         16, 128, 128, 8, 64, 64, 8, 4>
            <<<64, 256, 0, stream>>>(y1b, ew2, eb2, z + nb * ZSTR);
    }

    // VQ over the full z tensor.
    vq_init_kernel<<<1, 64, 0, stream>>>(stats);
    vq_kernel<<<NPOS / 256, 256, 0, stream>>>(z, emb, epo, zmix,
                                              out + OUTN + 1, stats);
    vq_finalize_kernel<<<1, 32, 0, stream>>>(stats, csz, epo, out);

    // Decoder, batch-chunked so decT2 reads y3 out of L2 (33.5 MB chunk).
    for (int nb = 0; nb < BATCH; ++nb) {
        // decT1: ConvT 8->32, s2 p1 op1, ReLU. tiles = 16*16*8 = 2048 -> 512
        conv_wmma_kernel<8, 8, 32, 2, 1, true, 0, true,
                         8, 64, 64, 16, 128, 128, 8, 4>
            <<<512, 256, 0, stream>>>(zmix + nb * ZSTR, dw1, db1, y3b);
        // decT2: ConvT 32->3, s1 p1 (= flipped conv), ReLU.
        // tiles = 16*32*8 = 4096 -> 1024 blocks of 4 waves
        conv_wmma_kernel<32, 32, 3, 1, 1, false, 0, true,
                         16, 128, 128, 16, 128, 128, 4, 4>
            <<<1024, 128, 0, stream>>>(y3b, dw2, db2, out + nb * XSTR);
    }
}